// GCN_46428596469830
// MI455X (gfx1250) — compile-verified
//
#include <hip/hip_runtime.h>
#include <math.h>

typedef __attribute__((ext_vector_type(2))) float v2f;
typedef __attribute__((ext_vector_type(8))) float v8f;

#define IN_C 128
#define LOG2C 7

// ---------------------------------------------------------------------------
// Small elementwise helpers
// ---------------------------------------------------------------------------
__global__ void k_set_const(float* __restrict__ p, float v, int n) {
    int i = blockIdx.x * blockDim.x + threadIdx.x;
    if (i < n) p[i] = v;
}

__global__ void k_count_deg(const int* __restrict__ dst, float* __restrict__ deg, int nE) {
    int i = blockIdx.x * blockDim.x + threadIdx.x;
    if (i < nE) atomicAdd(&deg[dst[i]], 1.0f);
}

__global__ void k_inv_sqrt(float* __restrict__ p, int n) {
    int i = blockIdx.x * blockDim.x + threadIdx.x;
    if (i < n) p[i] = 1.0f / sqrtf(p[i]);
}

// ---------------------------------------------------------------------------
// fp32 WMMA GEMM: C[M x 128] = A[M x 128] * W[128 x 128]   (both row-major)
// One wave computes a 16(M) x 128(N) strip: 8 accumulators of 16x16,
// built from 32 k-steps of V_WMMA_F32_16X16X4_F32. A fragment reused 8x.
// Fragment layouts per CDNA5 ISA 7.12.2 (32-bit A 16x4 / C 16x16).
// ---------------------------------------------------------------------------
__global__ __launch_bounds__(128) void k_gemm_wmma_f32(
    const float* __restrict__ A, const float* __restrict__ W,
    float* __restrict__ C, int mTiles)
{
    const int wavesPerBlock = blockDim.x >> 5;
    const int wave = blockIdx.x * wavesPerBlock + (threadIdx.x >> 5);
    if (wave >= mTiles) return;                 // wave-uniform: EXEC stays all-1s

    const int lane = threadIdx.x & 31;
    const int row  = lane & 15;                 // M index within tile (A), N index (B/C)
    const int hi   = lane >> 4;                 // half-wave select
    const int koff = hi * 2;                    // K sub-offset for A/B fragments
    const int m0   = wave * 16;

    v8f acc[8] = {};                            // 8 x (16x16 f32) accumulators

    const float* __restrict__ arow = A + (size_t)(m0 + row) * IN_C;

    for (int k = 0; k < IN_C; k += 4) {
        v2f a;
        a.x = arow[k + koff];                   // A[m0+row][k+koff]
        a.y = arow[k + koff + 1];               // A[m0+row][k+koff+1]
        const float* __restrict__ wbase = W + (size_t)(k + koff) * IN_C + row;
#pragma unroll
        for (int t = 0; t < 8; ++t) {
            v2f b;
            b.x = wbase[t * 16];                // W[k+koff  ][16t+row]
            b.y = wbase[t * 16 + IN_C];         // W[k+koff+1][16t+row]
            acc[t] = __builtin_amdgcn_wmma_f32_16x16x4_f32(
                false, a, false, b, (short)0, acc[t], false, false);
        }
    }

    // C/D layout: lane(l), vgpr(r) -> C[m0 + r + 8*hi][16t + (l&15)]
    float* __restrict__ cbase = C + (size_t)(m0 + hi * 8) * IN_C + row;
#pragma unroll
    for (int t = 0; t < 8; ++t) {
#pragma unroll
        for (int r = 0; r < 8; ++r) {
            cbase[(size_t)r * IN_C + t * 16] = acc[t][r];
        }
    }
}

// ---------------------------------------------------------------------------
// Edge aggregation: one wave per edge. Lane L owns channels 4L..4L+3.
// agg[dst] += h[src] * invsqrt(deg[src]) * invsqrt(deg[dst])
// h and agg are both L2-resident (51 MB each vs 192 MB L2), so atomics
// resolve in L2.
// ---------------------------------------------------------------------------
__global__ __launch_bounds__(256) void k_aggregate(
    const float* __restrict__ h,
    const int* __restrict__ src, const int* __restrict__ dst,
    const float* __restrict__ inv_sqrt,
    float* __restrict__ agg, int nE)
{
    const int wave = blockIdx.x * (blockDim.x >> 5) + (threadIdx.x >> 5);
    if (wave >= nE) return;
    const int lane = threadIdx.x & 31;

    const int s = src[wave];
    const int d = dst[wave];
    const float coef = inv_sqrt[s] * inv_sqrt[d];

    const float4 hv = ((const float4*)(h + (size_t)s * IN_C))[lane];
    float* ap = agg + (size_t)d * IN_C + lane * 4;
    atomicAdd(ap + 0, hv.x * coef);
    atomicAdd(ap + 1, hv.y * coef);
    atomicAdd(ap + 2, hv.z * coef);
    atomicAdd(ap + 3, hv.w * coef);
}

// ---------------------------------------------------------------------------
// out[i] = (agg[i] + h[i]*inv_deg[n] + bias[c])  [ReLU optional]; in-place OK
// ---------------------------------------------------------------------------
__global__ void k_finalize(
    const float* __restrict__ agg, const float* __restrict__ h,
    const float* __restrict__ inv_sqrt, const float* __restrict__ bias,
    float* __restrict__ out, int total, int do_relu)
{
    int i = blockIdx.x * blockDim.x + threadIdx.x;
    if (i >= total) return;
    int n = i >> LOG2C;
    int c = i & (IN_C - 1);
    float is = inv_sqrt[n];
    float v = agg[i] + h[i] * is * is + bias[c];
    if (do_relu) v = fmaxf(v, 0.0f);
    out[i] = v;
}

// ---------------------------------------------------------------------------
extern "C" void kernel_launch(void* const* d_in, const int* in_sizes, int n_in,
                              void* d_out, int out_size, void* d_ws, size_t ws_size,
                              hipStream_t stream)
{
    const float* x   = (const float*)d_in[0];
    const int*   ei  = (const int*)d_in[1];     // edge_index [2, E]
    const float* W1  = (const float*)d_in[2];
    const float* b1  = (const float*)d_in[3];
    const float* W2  = (const float*)d_in[4];
    const float* b2  = (const float*)d_in[5];

    const int N = in_sizes[0] / IN_C;
    const int E = in_sizes[1] / 2;
    const int* src = ei;
    const int* dst = ei + E;

    const int total = N * IN_C;

    // Workspace layout: inv_sqrt [N] (doubles as deg) | hbuf [N*128]
    float* inv_sqrt = (float*)d_ws;
    float* hbuf     = inv_sqrt + N;             // N*4 bytes offset (16B aligned for N%4==0)
    float* out      = (float*)d_out;            // doubles as aggregation accumulator

    const int T256 = 256;
    const int mTiles = (N + 15) / 16;           // N=100000 -> 6250 exact
    dim3 gemmGrid((mTiles + 3) / 4);            // 4 waves / 128-thread block
    dim3 gemmBlock(128);
    dim3 aggGrid((E + 7) / 8);                  // 8 waves / 256-thread block
    dim3 aggBlock(256);

    // --- degrees: deg = 1 (self-loop) + scatter count; then 1/sqrt ---
    k_set_const<<<(N + T256 - 1) / T256, T256, 0, stream>>>(inv_sqrt, 1.0f, N);
    k_count_deg<<<(E + T256 - 1) / T256, T256, 0, stream>>>(dst, inv_sqrt, E);
    k_inv_sqrt<<<(N + T256 - 1) / T256, T256, 0, stream>>>(inv_sqrt, N);

    // --- layer 1: h = x @ W1 ; agg ; finalize(+b1, ReLU) -> d_out holds x2 ---
    k_gemm_wmma_f32<<<gemmGrid, gemmBlock, 0, stream>>>(x, W1, hbuf, mTiles);
    k_set_const<<<(total + T256 - 1) / T256, T256, 0, stream>>>(out, 0.0f, total);
    k_aggregate<<<aggGrid, aggBlock, 0, stream>>>(hbuf, src, dst, inv_sqrt, out, E);
    k_finalize<<<(total + T256 - 1) / T256, T256, 0, stream>>>(
        out, hbuf, inv_sqrt, b1, out, total, /*relu=*/1);

    // --- layer 2: h = x2 @ W2 ; agg ; finalize(+b2) -> d_out ---
    k_gemm_wmma_f32<<<gemmGrid, gemmBlock, 0, stream>>>(out, W2, hbuf, mTiles);
    k_set_const<<<(total + T256 - 1) / T256, T256, 0, stream>>>(out, 0.0f, total);
    k_aggregate<<<aggGrid, aggBlock, 0, stream>>>(hbuf, src, dst, inv_sqrt, out, E);
    k_finalize<<<(total + T256 - 1) / T256, T256, 0, stream>>>(
        out, hbuf, inv_sqrt, b2, out, total, /*relu=*/0);
}